// OptimizedDASPACE_24421184045389
// MI455X (gfx1250) — compile-verified
//
#include <hip/hip_runtime.h>
#include <math.h>

#define T_LEN 88200
#define NWAVE 32
#define SEG   2784   // waves 0..30 take 2784 (=87*32), wave 31 takes 1896 (=59*32+8)

#ifndef __has_builtin
#define __has_builtin(x) 0
#endif

#if __has_builtin(__builtin_amdgcn_global_load_async_to_lds_b32)
#define HAVE_ASYNC_LDS 1
#else
#define HAVE_ASYNC_LDS 0
#endif

typedef __attribute__((address_space(1))) int gas_int;  // prints as "__device__ int *"
typedef __attribute__((address_space(3))) int las_int;  // prints as "__shared__ int *"

// Per-alpha, per-lane scan constants.
//   a = alpha, om = 1-alpha
//   c2/c4/c8 = alpha^min(r+1, sigma) for DPP steps sigma=2,4,8 (r = lane&15); step 1 uses a
//   cx  = alpha^(r+1)    (cross-row compose with lane 15)
//   Bf  = alpha^(lane+1) (full inclusive B for applying the carry)
//   B32 = alpha^32       (full-tile transfer B)
struct ACons { float a, om, c2, c4, c8, cx, Bf, B32; };

__device__ __forceinline__ float readlane_f(float v, int l) {
  return __int_as_float(__builtin_amdgcn_readlane(__float_as_int(v), l));
}

template<int CTRL>
__device__ __forceinline__ float mov_dpp_f(float src, float oldv) {
  return __int_as_float(__builtin_amdgcn_update_dpp(
      __float_as_int(oldv), __float_as_int(src), CTRL, 0xF, 0xF, false));
}

template<bool ISMAX>
__device__ __forceinline__ float selmm(float x, float y) {
  return ISMAX ? fmaxf(x, y) : fminf(x, y);
}

#if HAVE_ASYNC_LDS
__device__ __forceinline__ void async_stage(const float* gsrc, float* ldst) {
  // GLOBAL_LOAD_ASYNC_TO_LDS_B32: one dword per lane, tracked by ASYNCcnt.
  __builtin_amdgcn_global_load_async_to_lds_b32(
      (gas_int*)(void*)const_cast<float*>(gsrc),
      (las_int*)(void*)ldst, 0, 0);
}
__device__ __forceinline__ void wait_async0() {
#if __has_builtin(__builtin_amdgcn_s_wait_asynccnt)
  __builtin_amdgcn_s_wait_asynccnt(0);
#else
  asm volatile("s_wait_asynccnt 0x0" ::: "memory");
#endif
}
__device__ __forceinline__ void fence_ds() {
  asm volatile("s_wait_dscnt 0x0" ::: "memory");
}
#endif

// Inclusive wave32 scan of affine-max/min maps (C, A) with deterministic B.
// Map: f(y) = max_or_min(C, A + B*y). Composition (newer=2, older=1):
//   C' = mm(C2, A2 + B2*C1); A' = A2 + B2*A1.
template<bool ISMAX>
__device__ __forceinline__ void core_scan(float& C, float& A, const ACons& k, int lane) {
  const float IDC = ISMAX ? -3.0e38f : 3.0e38f;  // identity C; A identity = 0
  float Cp, Ap, nC;
  // sigma = 1
  Cp = mov_dpp_f<0x111>(C, IDC); Ap = mov_dpp_f<0x111>(A, 0.0f);
  nC = selmm<ISMAX>(C, fmaf(k.a,  Cp, A)); A = fmaf(k.a,  Ap, A); C = nC;
  // sigma = 2
  Cp = mov_dpp_f<0x112>(C, IDC); Ap = mov_dpp_f<0x112>(A, 0.0f);
  nC = selmm<ISMAX>(C, fmaf(k.c2, Cp, A)); A = fmaf(k.c2, Ap, A); C = nC;
  // sigma = 4
  Cp = mov_dpp_f<0x114>(C, IDC); Ap = mov_dpp_f<0x114>(A, 0.0f);
  nC = selmm<ISMAX>(C, fmaf(k.c4, Cp, A)); A = fmaf(k.c4, Ap, A); C = nC;
  // sigma = 8  (row-local inclusive complete after this)
  Cp = mov_dpp_f<0x118>(C, IDC); Ap = mov_dpp_f<0x118>(A, 0.0f);
  nC = selmm<ISMAX>(C, fmaf(k.c8, Cp, A)); A = fmaf(k.c8, Ap, A); C = nC;
  // cross-row fixup: compose lanes 16..31 with lane 15's row-0 inclusive map
  float C15 = readlane_f(C, 15);
  float A15 = readlane_f(A, 15);
  if (lane >= 16) {
    nC = selmm<ISMAX>(C, fmaf(k.cx, C15, A));
    A  = fmaf(k.cx, A15, A);
    C  = nC;
  }
}

// PASS 1: compose decay(|x|) segment transfer                   -> (gC,gA,gB) max-form
// PASS 2: apply decay (carry cd), compose attack(d)             -> (gC,gA,gB) min-form
// PASS 3: apply decay+attack, et=relu(d-a-nu), compose decay(et)-> max-form
// PASS 4: apply everything, out = x * et / (e + reg)
template<int PASS>
__device__ void do_pass(const float* __restrict__ Xrow, float* __restrict__ Orow,
                        int segLen, int lane, int w,
                        const ACons& kD, const ACons& kA,
                        float nuAmp, float regAmp,
                        float cd, float ca, float ce,
                        const float* pwD, const float* pwA,
                        float* gC, float* gA, float* gB,
                        float* stg /* per-wave [2][32] staging buffer */) {
  float sC = (PASS == 2) ? 3.0e38f : -3.0e38f;  // identity transfer
  float sA = 0.0f, sB = 1.0f;

  auto tile = [&](int t, int valid, float xs) {
    const int off = (t << 5) + lane;
    float xa = fabsf(xs);

    // decay env on |x|
    float C = xa, A = kD.om * xa;
    core_scan<true>(C, A, kD, lane);
    if (PASS == 1) {
      float Ct = readlane_f(C, valid - 1), At = readlane_f(A, valid - 1);
      float Bt = (valid == 32) ? kD.B32 : pwD[valid];
      sC = fmaxf(Ct, fmaf(Bt, sC, At)); sA = fmaf(Bt, sA, At); sB *= Bt;
      return;
    }
    float d = fmaxf(C, fmaf(kD.Bf, cd, A));
    cd = readlane_f(d, 31);

    // attack env on d
    C = d; A = kA.om * d;
    core_scan<false>(C, A, kA, lane);
    if (PASS == 2) {
      float Ct = readlane_f(C, valid - 1), At = readlane_f(A, valid - 1);
      float Bt = (valid == 32) ? kA.B32 : pwA[valid];
      sC = fminf(Ct, fmaf(Bt, sC, At)); sA = fmaf(Bt, sA, At); sB *= Bt;
      return;
    }
    float a = fminf(C, fmaf(kA.Bf, ca, A));
    ca = readlane_f(a, 31);

    float et = fmaxf(d - a - nuAmp, 0.0f);

    // decay env on et
    C = et; A = kD.om * et;
    core_scan<true>(C, A, kD, lane);
    if (PASS == 3) {
      float Ct = readlane_f(C, valid - 1), At = readlane_f(A, valid - 1);
      float Bt = (valid == 32) ? kD.B32 : pwD[valid];
      sC = fmaxf(Ct, fmaf(Bt, sC, At)); sA = fmaf(Bt, sA, At); sB *= Bt;
      return;
    }
    float e = fmaxf(C, fmaf(kD.Bf, ce, A));
    ce = readlane_f(e, 31);

    float mod = et / (e + regAmp);
    if (valid == 32 || lane < valid) Orow[off] = xs * mod;
  };

  const int nFull = segLen >> 5;
  const int rem   = segLen & 31;

#if HAVE_ASYNC_LDS
  // Double-buffered async global->LDS pipeline: tile t+1 streams into LDS via
  // the async engine while the DPP scan chain processes tile t.
  async_stage(Xrow + lane, stg + lane);  // tile 0 -> buffer 0
  for (int t = 0; t < nFull; ++t) {
    wait_async0();                                    // tile t landed (in-order)
    float xs = stg[((t & 1) << 5) + lane];            // ds_load
    fence_ds();                                       // buffer provably consumed
    if (t + 1 < nFull)
      async_stage(Xrow + ((t + 1) << 5) + lane, stg + (((t + 1) & 1) << 5) + lane);
    if (PASS == 1) __builtin_prefetch(Xrow + (t << 5) + lane + 1024, 0, 1);
    tile(t, 32, xs);
  }
#else
  for (int t = 0; t < nFull; ++t) {
    float xs = Xrow[(t << 5) + lane];
    if (PASS == 1) __builtin_prefetch(Xrow + (t << 5) + lane + 1024, 0, 1);
    tile(t, 32, xs);
  }
  (void)stg;
#endif
  if (rem) {
    const int off = (nFull << 5) + lane;
    float xs = (lane < rem) ? Xrow[off] : 0.0f;  // predicated: async must not go OOB
    tile(nFull, rem, xs);
  }

  if (PASS != 4 && lane == 0) { gC[w] = sC; gA[w] = sA; gB[w] = sB; }
}

__global__ void __launch_bounds__(1024, 1)
tce_scan_kernel(const float* __restrict__ X, float* __restrict__ O,
                const float* __restrict__ pTauA, const float* __restrict__ pTauD,
                const float* __restrict__ pNu, const float* __restrict__ pReg) {
  __shared__ float powD[33], powA[33];
  __shared__ float gC[NWAVE], gA[NWAVE], gB[NWAVE];
  __shared__ float carryD[NWAVE], carryA[NWAVE], carryE[NWAVE];
  __shared__ float par[4];
  __shared__ float sStage[NWAVE * 64];  // 2 x 32 floats per wave (async staging)

  const int tid  = threadIdx.x;
  const int lane = tid & 31;
  const int w    = tid >> 5;
  const int row  = blockIdx.x;

  if (tid == 0) {
    float tD = fminf(fmaxf(pTauD[0], 1.0f), 100.0f);  // clip(1,100) then clip(0.1,1000) = clip(1,100)
    float tA = fminf(fmaxf(pTauA[0], 1.0f), 100.0f);
    float aD = expf(-1000.0f / (tD * 44100.0f));
    float aA = expf(-1000.0f / (tA * 44100.0f));
    par[0] = aD;
    par[1] = aA;
    par[2] = powf(10.0f, fminf(fmaxf(pNu[0],  -60.0f),   0.0f) * 0.05f);
    par[3] = powf(10.0f, fminf(fmaxf(pReg[0], -120.0f), -60.0f) * 0.05f);
    powD[0] = 1.0f; powA[0] = 1.0f;
    for (int k = 1; k <= 32; ++k) {
      powD[k] = powD[k - 1] * aD;
      powA[k] = powA[k - 1] * aA;
    }
  }
  __syncthreads();

  const float aD = par[0], aA = par[1], nuAmp = par[2], regAmp = par[3];
  const int r  = lane & 15;
  const int l1 = r + 1;
  ACons kD = { aD, 1.0f - aD, powD[l1 < 2 ? l1 : 2], powD[l1 < 4 ? l1 : 4],
               powD[l1 < 8 ? l1 : 8], powD[l1], powD[lane + 1], powD[32] };
  ACons kA = { aA, 1.0f - aA, powA[l1 < 2 ? l1 : 2], powA[l1 < 4 ? l1 : 4],
               powA[l1 < 8 ? l1 : 8], powA[l1], powA[lane + 1], powA[32] };

  const int segStart = w * SEG;
  const int segLen   = (w == NWAVE - 1) ? (T_LEN - SEG * (NWAVE - 1)) : SEG;
  const float* Xrow = X + (size_t)row * T_LEN + segStart;
  float*       Orow = O + (size_t)row * T_LEN + segStart;
  float*       stg  = sStage + w * 64;

  // Pass 1: decay segment transfer functions
  do_pass<1>(Xrow, Orow, segLen, lane, w, kD, kA, nuAmp, regAmp,
             0.0f, 0.0f, 0.0f, powD, powA, gC, gA, gB, stg);
  __syncthreads();
  if (tid == 0) {
    float y = 0.0f;
    for (int i = 0; i < NWAVE; ++i) { carryD[i] = y; y = fmaxf(gC[i], fmaf(gB[i], y, gA[i])); }
  }
  __syncthreads();

  // Pass 2: attack segment transfer functions
  do_pass<2>(Xrow, Orow, segLen, lane, w, kD, kA, nuAmp, regAmp,
             carryD[w], 0.0f, 0.0f, powD, powA, gC, gA, gB, stg);
  __syncthreads();
  if (tid == 0) {
    float y = 0.0f;
    for (int i = 0; i < NWAVE; ++i) { carryA[i] = y; y = fminf(gC[i], fmaf(gB[i], y, gA[i])); }
  }
  __syncthreads();

  // Pass 3: decay(et) segment transfer functions
  do_pass<3>(Xrow, Orow, segLen, lane, w, kD, kA, nuAmp, regAmp,
             carryD[w], carryA[w], 0.0f, powD, powA, gC, gA, gB, stg);
  __syncthreads();
  if (tid == 0) {
    float y = 0.0f;
    for (int i = 0; i < NWAVE; ++i) { carryE[i] = y; y = fmaxf(gC[i], fmaf(gB[i], y, gA[i])); }
  }
  __syncthreads();

  // Pass 4: apply everything and write out
  do_pass<4>(Xrow, Orow, segLen, lane, w, kD, kA, nuAmp, regAmp,
             carryD[w], carryA[w], carryE[w], powD, powA, gC, gA, gB, stg);
}

extern "C" void kernel_launch(void* const* d_in, const int* in_sizes, int n_in,
                              void* d_out, int out_size, void* d_ws, size_t ws_size,
                              hipStream_t stream) {
  (void)in_sizes; (void)n_in; (void)d_ws; (void)ws_size;
  const float* X    = (const float*)d_in[0];
  const float* tauA = (const float*)d_in[1];
  const float* tauD = (const float*)d_in[2];
  const float* nu   = (const float*)d_in[3];
  const float* reg  = (const float*)d_in[4];
  float* O = (float*)d_out;
  const int rows = out_size / T_LEN;  // 8*32 = 256
  tce_scan_kernel<<<rows, 1024, 0, stream>>>(X, O, tauA, tauD, nu, reg);
}